// GraphConvolutionalAutoencoder_76124000354640
// MI455X (gfx1250) — compile-verified
//
#include <hip/hip_runtime.h>
#include <hip/hip_bf16.h>

// ---------------------------------------------------------------------------
// GraphConvolutionalAutoencoder on MI455X (gfx1250, wave32)
//
// Shapes: B=16, N=2048, C=32, K=5, E=524288, FFN=512, BOT=64, NL=2
// Nodes NN = B*N = 32768.  All data fp32 -> V_WMMA_F32_16X16X4_F32 keeps
// full fp32 precision on the tensor path.
//
// HBM floor: fc_enc1_w + fc_dec2_w = 268 MB streamed once -> ~11.5us at
// 23.3 TB/s.  Split-K (64 chunks) + fp32 atomic accumulation fixes the
// occupancy problem of the M=16 GEMMs (32 tiles -> 2048 waves).  The GEMM
// inner loop is unrolled 4x (16 k-values): 8 batched loads -> one wait ->
// 4 chained WMMAs, so each wave keeps multiple loads in flight instead of
// stalling once per WMMA (critical for saturating the weight stream).
// Edge gathers hit L2 (xg = 21 MB << 192 MB L2); scatter via
// global_atomic_add_f32 into a 4 MB L2-resident agg buffer.
// ---------------------------------------------------------------------------

#define NNODES 32768     // B*N
#define CCH    32        // C
#define KK     5         // gaussian kernels
#define NEDGE  524288    // E
#define XGW    160       // C*K

typedef float v2f __attribute__((ext_vector_type(2)));
typedef float v8f __attribute__((ext_vector_type(8)));

__device__ __forceinline__ float elu1(float v) {
  return v > 0.0f ? v : (expf(v) - 1.0f);
}

// ---------------------------------------------------------------------------
// Generic fp32 WMMA GEMM:  C[M,N] (+)= A[M,Kd] @ B[Kd,N]
// One wave32 per 16x16 C tile, K stepped by 4 via V_WMMA_F32_16X16X4_F32.
// Fragment layouts per CDNA5 ISA 7.12.2:
//   A 16x4 : lane l<16 -> (m=l, k..k+1), lane l>=16 -> (m=l-16, k+2..k+3)
//   B 4x16 : lane l<16 -> (k..k+1, n=l), lane l>=16 -> (k+2..k+3, n=l-16)
//   C 16x16: lane l -> col (l&15), rows (l<16 ? 0..7 : 8..15)
// numChunks==1  -> plain store;  else fp32 atomicAdd into pre-zeroed C.
// ---------------------------------------------------------------------------
__global__ __launch_bounds__(128)
void wmma_gemm_f32(const float* __restrict__ A, const float* __restrict__ B,
                   float* __restrict__ C, int M, int Kd, int N,
                   int kChunk, int numChunks) {
  const int lane  = threadIdx.x & 31;
  const int wave  = threadIdx.x >> 5;
  const int tilesN = N >> 4;
  const int tilesTotal = (M >> 4) * tilesN;
  const int tile = blockIdx.x * (blockDim.x >> 5) + wave;
  if (tile >= tilesTotal) return;

  const int tm = (tile / tilesN) << 4;   // tile row base in C
  const int tn = (tile % tilesN) << 4;   // tile col base in C
  const int k0 = blockIdx.y * kChunk;
  int k1 = k0 + kChunk; if (k1 > Kd) k1 = Kd;

  const int half = lane >> 4;            // 0: lanes 0-15, 1: lanes 16-31
  const int l    = lane & 15;

  v8f acc = {};
  const float* arow = A + (size_t)(tm + l) * Kd + (half << 1);        // (m, k+2h)
  const float* bcol = B + (size_t)(half << 1) * N + tn + l;           // (k+2h, n)

  int k = k0;
  // main loop: 16 k-values per iteration -> 8 loads batched, 4 WMMAs chained.
  for (; k + 16 <= k1; k += 16) {
    v2f a[4], b[4];
#pragma unroll
    for (int u = 0; u < 4; ++u) {
      const float* ap = arow + k + 4 * u;
      a[u].x = ap[0];
      a[u].y = ap[1];
      const float* bp = bcol + (size_t)(k + 4 * u) * N;
      b[u].x = bp[0];
      b[u].y = bp[N];
    }
#pragma unroll
    for (int u = 0; u < 4; ++u) {
      acc = __builtin_amdgcn_wmma_f32_16x16x4_f32(
          false, a[u], false, b[u], (short)0, acc, false, false);
    }
  }
  // tail (unused for this model's K dims, kept for generality)
  for (; k < k1; k += 4) {
    v2f a, b;
    const float* ap = arow + k;
    a.x = ap[0];
    a.y = ap[1];
    const float* bp = bcol + (size_t)k * N;
    b.x = bp[0];
    b.y = bp[N];
    acc = __builtin_amdgcn_wmma_f32_16x16x4_f32(
        false, a, false, b, (short)0, acc, false, false);
  }

  float* cp = C + (size_t)(tm + (half << 3)) * N + tn + l;
  if (numChunks == 1) {
#pragma unroll
    for (int i = 0; i < 8; ++i) cp[(size_t)i * N] = acc[i];
  } else {
#pragma unroll
    for (int i = 0; i < 8; ++i) atomicAdd(&cp[(size_t)i * N], acc[i]);
  }
}

// ---------------------------------------------------------------------------
// Gaussian edge weights: gauss[e,k] = exp(-0.5*(ea[e]-mu[k])^2/(eps+sig[k]^2))
// ---------------------------------------------------------------------------
__global__ void gauss_kernel(const float* __restrict__ ea,
                             const float* __restrict__ mu,
                             const float* __restrict__ sigma,
                             float* __restrict__ gauss, int E) {
  int e = blockIdx.x * blockDim.x + threadIdx.x;
  if (e >= E) return;
  float a = ea[e];
#pragma unroll
  for (int k = 0; k < KK; ++k) {
    float d = a - mu[k];
    float s = sigma[k];
    gauss[(size_t)e * KK + k] = expf(-0.5f * d * d / (1e-15f + s * s));
  }
}

// in-degree counts (depends only on dst -> computed once, reused 4x)
__global__ void cnt_kernel(const int* __restrict__ dst,
                           float* __restrict__ cnt, int E) {
  int e = blockIdx.x * blockDim.x + threadIdx.x;
  if (e < E) atomicAdd(&cnt[dst[e]], 1.0f);
}

// ---------------------------------------------------------------------------
// Edge message + scatter: one wave32 per edge, lane = output channel.
//   msg[m] = sum_k gauss[e,k] * xg[src[e], k*32 + m];  atomicAdd agg[dst,m]
// xg rows (640 B) gather from L2; agg (4 MB) atomics resolve in L2.
// ---------------------------------------------------------------------------
__global__ __launch_bounds__(256)
void edge_msg_kernel(const float* __restrict__ xg,
                     const float* __restrict__ gauss,
                     const int* __restrict__ src,
                     const int* __restrict__ dst,
                     float* __restrict__ agg, int E) {
  int e = (int)((blockIdx.x * blockDim.x + threadIdx.x) >> 5);
  int lane = threadIdx.x & 31;
  if (e >= E) return;
  const float* xr = xg + (size_t)src[e] * XGW;
  const float* gw = gauss + (size_t)e * KK;
  float m = 0.0f;
#pragma unroll
  for (int k = 0; k < KK; ++k) m = fmaf(gw[k], xr[k * CCH + lane], m);
  atomicAdd(&agg[(size_t)dst[e] * CCH + lane], m);
}

// ---------------------------------------------------------------------------
// GMMConv epilogue: out = [elu]( agg/max(cnt,1) + x@root + bias + h )
// ---------------------------------------------------------------------------
__global__ void conv_combine_kernel(const float* __restrict__ agg,
                                    const float* __restrict__ cnt,
                                    const float* __restrict__ rb,
                                    const float* __restrict__ bias,
                                    const float* __restrict__ h,
                                    float* __restrict__ out,
                                    int n, int doAct) {
  int i = blockIdx.x * blockDim.x + threadIdx.x;
  if (i >= n) return;
  int node = i >> 5;          // C == 32
  int ch   = i & 31;
  float c = fmaxf(cnt[node], 1.0f);
  float v = agg[i] / c + rb[i] + bias[ch] + h[i];
  out[i] = doAct ? elu1(v) : v;
}

// FFN epilogue: out[i] = [elu]( in[i] + bias[i % N] )   (in-place safe)
__global__ void bias_act_kernel(const float* __restrict__ in,
                                const float* __restrict__ bias,
                                float* __restrict__ out,
                                int n, int N, int doAct) {
  int i = blockIdx.x * blockDim.x + threadIdx.x;
  if (i >= n) return;
  float v = in[i] + bias[i % N];
  out[i] = doAct ? elu1(v) : v;
}

// ---------------------------------------------------------------------------

struct ConvP { const float *g, *mu, *sigma, *root, *bias; };

static inline void launch_gemm(const float* A, const float* B, float* C,
                               int M, int Kd, int N, int kChunk, int numChunks,
                               hipStream_t s) {
  int tiles = (M >> 4) * (N >> 4);
  dim3 grid((tiles + 3) / 4, numChunks);
  wmma_gemm_f32<<<grid, 128, 0, s>>>(A, B, C, M, Kd, N, kChunk, numChunks);
}

extern "C" void kernel_launch(void* const* d_in, const int* in_sizes, int n_in,
                              void* d_out, int out_size, void* d_ws, size_t ws_size,
                              hipStream_t stream) {
  const float* x_in  = (const float*)d_in[0];          // [32768, 32]
  const int*   eidx  = (const int*)  d_in[1];          // [2, E]
  const float* eattr = (const float*)d_in[2];          // [E, 1]
  const int* src = eidx;                               // row 0
  const int* dst = eidx + NEDGE;                       // row 1

  // params in setup_inputs() insertion order; each conv dict: g,mu,sigma,root,bias
  int idx = 3;
  ConvP enc[2], dec[2];
  for (int i = 0; i < 2; ++i) {
    enc[i].g = (const float*)d_in[idx++]; enc[i].mu = (const float*)d_in[idx++];
    enc[i].sigma = (const float*)d_in[idx++]; enc[i].root = (const float*)d_in[idx++];
    enc[i].bias = (const float*)d_in[idx++];
  }
  for (int i = 0; i < 2; ++i) {
    dec[i].g = (const float*)d_in[idx++]; dec[i].mu = (const float*)d_in[idx++];
    dec[i].sigma = (const float*)d_in[idx++]; dec[i].root = (const float*)d_in[idx++];
    dec[i].bias = (const float*)d_in[idx++];
  }
  const float* fc1w = (const float*)d_in[idx++];  // [65536, 512]
  const float* fc1b = (const float*)d_in[idx++];  // [512]
  const float* fc2w = (const float*)d_in[idx++];  // [512, 64]
  const float* fc2b = (const float*)d_in[idx++];  // [64]
  const float* fc3w = (const float*)d_in[idx++];  // [64, 512]
  const float* fc3b = (const float*)d_in[idx++];  // [512]
  const float* fc4w = (const float*)d_in[idx++];  // [512, 65536]
  const float* fc4b = (const float*)d_in[idx++];  // [65536]

  // ---- workspace carve-up (floats) -------------------------------------
  float* W = (float*)d_ws;
  size_t o = 0;
  const size_t NC = (size_t)NNODES * CCH;                 // 1,048,576
  float* bufA  = W + o; o += NC;
  float* bufB  = W + o; o += NC;
  float* xg    = W + o; o += (size_t)NNODES * XGW;        // 5,242,880
  float* agg   = W + o; o += NC;
  float* rb    = W + o; o += NC;
  float* cnt   = W + o; o += NNODES;
  float* gauss = W + o; o += (size_t)NEDGE * KK;          // 2,621,440
  float* z1    = W + o; o += 16 * 512;
  float* z2    = W + o; o += 16 * 64;
  float* dd1   = W + o; o += 16 * 512;
  float* xpre  = W + o; o += NC;                          // total ~13.2M floats

  const int ethreads = 256;
  const int eblocks  = (NEDGE + ethreads - 1) / ethreads;
  const int ncblocks = (int)((NC + 255) / 256);

  // in-degree counts: once, reused by all 4 conv layers
  hipMemsetAsync(cnt, 0, NNODES * sizeof(float), stream);
  cnt_kernel<<<eblocks, ethreads, 0, stream>>>(dst, cnt, NEDGE);

  // one GMMConv layer
  auto conv = [&](const float* in, const float* h, float* out,
                  const ConvP& p, int doAct) {
    launch_gemm(in, p.g,    xg, NNODES, CCH, XGW, CCH, 1, stream);  // x@g
    launch_gemm(in, p.root, rb, NNODES, CCH, CCH, CCH, 1, stream);  // x@root
    gauss_kernel<<<eblocks, ethreads, 0, stream>>>(eattr, p.mu, p.sigma,
                                                   gauss, NEDGE);
    hipMemsetAsync(agg, 0, NC * sizeof(float), stream);
    edge_msg_kernel<<<(NEDGE * 32 + 255) / 256, 256, 0, stream>>>(
        xg, gauss, src, dst, agg, NEDGE);
    conv_combine_kernel<<<ncblocks, 256, 0, stream>>>(
        agg, cnt, rb, p.bias, h, out, (int)NC, doAct);
  };

  // ---- encoder convs (residual h = original x) -------------------------
  conv(x_in, x_in, bufB, enc[0], 1);
  conv(bufB, x_in, bufA, enc[1], 1);

  // ---- FFN bottleneck ([16, 65536] view of bufA) -----------------------
  // fc_enc1: split-K (64 chunks) -> atomic accumulate, then bias+ELU
  hipMemsetAsync(z1, 0, 16 * 512 * sizeof(float), stream);
  launch_gemm(bufA, fc1w, z1, 16, 65536, 512, 1024, 64, stream);
  bias_act_kernel<<<(16 * 512 + 255) / 256, 256, 0, stream>>>(
      z1, fc1b, z1, 16 * 512, 512, 1);
  // fc_enc2: split-K (8 chunks), bias only
  hipMemsetAsync(z2, 0, 16 * 64 * sizeof(float), stream);
  launch_gemm(z1, fc2w, z2, 16, 512, 64, 64, 8, stream);
  bias_act_kernel<<<(16 * 64 + 255) / 256, 256, 0, stream>>>(
      z2, fc2b, z2, 16 * 64, 64, 0);
  // fc_dec1: single chunk, bias+ELU
  launch_gemm(z2, fc3w, dd1, 16, 64, 512, 64, 1, stream);
  bias_act_kernel<<<(16 * 512 + 255) / 256, 256, 0, stream>>>(
      dd1, fc3b, dd1, 16 * 512, 512, 1);
  // fc_dec2: 4096 tiles -> plenty of waves, single chunk; bias+ELU -> bufB
  launch_gemm(dd1, fc4w, xpre, 16, 512, 65536, 512, 1, stream);
  bias_act_kernel<<<ncblocks, 256, 0, stream>>>(
      xpre, fc4b, bufB, (int)NC, 65536, 1);

  // ---- decoder convs (residual h = FFN output; last layer no act) ------
  conv(bufB, bufB, bufA, dec[0], 1);
  conv(bufA, bufB, (float*)d_out, dec[1], 0);
}